// GNN_80255758893109
// MI455X (gfx1250) — compile-verified
//
#include <hip/hip_runtime.h>
#include <hip/hip_bf16.h>
#include <math.h>

#define EPS 1e-5f

typedef float v2f __attribute__((ext_vector_type(2)));
typedef float v8f __attribute__((ext_vector_type(8)));

// ---------------------------------------------------------------------------
// CSR construction: histogram, scan, placement
// ---------------------------------------------------------------------------
__global__ void fill_zero_i32(int* p, int n) {
    int i = blockIdx.x * blockDim.x + threadIdx.x;
    if (i < n) p[i] = 0;
}

__global__ void cnt_accum(const int* __restrict__ dst, int* __restrict__ cnt, int E) {
    int e = blockIdx.x * blockDim.x + threadIdx.x;
    if (e < E) atomicAdd(&cnt[dst[e]], 1);
}

__global__ void dinv_from_cnt(const int* __restrict__ cnt, float* __restrict__ dinv, int n) {
    int i = blockIdx.x * blockDim.x + threadIdx.x;
    if (i < n) dinv[i] = rsqrtf((float)cnt[i] + 1.0f);
}

// Hillis-Steele block scan: out = exclusive scan of cnt within block,
// blockSums[b] = block total.
__global__ void scan_block(const int* __restrict__ cnt, int* __restrict__ out,
                           int* __restrict__ blockSums, int n) {
    __shared__ int s[256];
    int tid = threadIdx.x;
    int i = blockIdx.x * 256 + tid;
    int v = (i < n) ? cnt[i] : 0;
    s[tid] = v;
    __syncthreads();
    for (int off = 1; off < 256; off <<= 1) {
        int t = (tid >= off) ? s[tid - off] : 0;
        __syncthreads();
        s[tid] += t;
        __syncthreads();
    }
    if (i < n) out[i] = s[tid] - v;                 // exclusive
    if (tid == 255) blockSums[blockIdx.x] = s[255]; // total
}

// Single-block exclusive scan of block sums (nb <= 512).
__global__ void scan_sums(int* __restrict__ blockSums, int nb) {
    __shared__ int s[512];
    int tid = threadIdx.x;
    int v = (tid < nb) ? blockSums[tid] : 0;
    s[tid] = v;
    __syncthreads();
    for (int off = 1; off < 512; off <<= 1) {
        int t = (tid >= off) ? s[tid - off] : 0;
        __syncthreads();
        s[tid] += t;
        __syncthreads();
    }
    if (tid < nb) blockSums[tid] = s[tid] - v;      // exclusive
}

__global__ void scan_add_cursor(int* __restrict__ rowStart, const int* __restrict__ blockSums,
                                int* __restrict__ cursor, int n) {
    int i = blockIdx.x * 256 + threadIdx.x;
    if (i < n) {
        int r = rowStart[i] + blockSums[blockIdx.x];
        rowStart[i] = r;
        cursor[i] = r;
    }
}

__global__ void csr_place(const int* __restrict__ src, const int* __restrict__ dst,
                          int* __restrict__ cursor, int* __restrict__ csrSrc, int E) {
    int e = blockIdx.x * blockDim.x + threadIdx.x;
    if (e < E) {
        int p = atomicAdd(&cursor[dst[e]], 1);
        csrSrc[p] = src[e];
    }
}

// ---------------------------------------------------------------------------
// fp32 WMMA GEMM:  H[N,F] = X[N,K] @ W[K,F]
// One block covers 32 rows; wave w covers output cols [16w,16w+16) for both
// 16-row tiles, reusing one B fragment per k-step across two WMMAs.
// A/B fragment layout per CDNA5 ISA 7.12.2:
//   lanes 0-15 : v0=K0, v1=K1 ; lanes 16-31 : v0=K2, v1=K3
// C/D: VGPR r -> row r (lanes 0-15) / row r+8 (lanes 16-31), col = lane&15.
// ---------------------------------------------------------------------------
__global__ void gcn_gemm_f32_wmma(const float* __restrict__ X,
                                  const float* __restrict__ W,
                                  float* __restrict__ H,
                                  int Nn, int K, int F) {
    extern __shared__ float lds[];           // 32 x (K+4) floats, padded rows
    const int ldK  = K + 4;
    const int tid  = threadIdx.x;
    const int lane = tid & 31;
    const int wave = tid >> 5;
    const int rowBase = blockIdx.x * 32;

    // Cooperative float4 staging of the 32 x K activation tile.
    const int kc = K >> 2;                   // float4 chunks per row
    for (int i = tid; i < 32 * kc; i += blockDim.x) {
        int r = i / kc;
        int c4 = (i - r * kc) << 2;
        int gr = rowBase + r;
        float4 v = (gr < Nn) ? *(const float4*)(X + (size_t)gr * K + c4)
                             : make_float4(0.f, 0.f, 0.f, 0.f);
        *(float4*)(lds + r * ldK + c4) = v;
    }
    __syncthreads();

    const int colBase = wave * 16;
    const int half = lane >> 4;
    const int l15  = lane & 15;

    v8f acc0 = {0.f, 0.f, 0.f, 0.f, 0.f, 0.f, 0.f, 0.f};
    v8f acc1 = {0.f, 0.f, 0.f, 0.f, 0.f, 0.f, 0.f, 0.f};

    for (int k = 0; k < K; k += 4) {
        const int ka = k + (half ? 2 : 0);
        v2f b, a0, a1;
        b.x = W[(size_t)ka       * F + colBase + l15];
        b.y = W[(size_t)(ka + 1) * F + colBase + l15];
        a0.x = lds[l15 * ldK + ka];
        a0.y = lds[l15 * ldK + ka + 1];
        a1.x = lds[(16 + l15) * ldK + ka];
        a1.y = lds[(16 + l15) * ldK + ka + 1];
        acc0 = __builtin_amdgcn_wmma_f32_16x16x4_f32(
                   false, a0, false, b, (short)0, acc0, false, false);
        acc1 = __builtin_amdgcn_wmma_f32_16x16x4_f32(
                   false, a1, false, b, (short)0, acc1, false, false);
    }

    const int rowOff0 = rowBase + (half ? 8 : 0);
#pragma unroll
    for (int r = 0; r < 8; ++r) {
        int gr = rowOff0 + r;
        if (gr < Nn) H[(size_t)gr * F + colBase + l15] = acc0[r];
        int gr1 = gr + 16;
        if (gr1 < Nn) H[(size_t)gr1 * F + colBase + l15] = acc1[r];
    }
}

// ---------------------------------------------------------------------------
// Fused CSR gather + self-loop + bias + BN + ReLU, F=128.
// One wave per node; lane handles a float4 feature chunk (32*4 = 128).
// h rows are L2-resident (51MB << 192MB) -> pure L2 loads, no atomics.
// ---------------------------------------------------------------------------
__global__ void gather_bn_relu_f128(const float* __restrict__ h,
                                    const int* __restrict__ rowStart,
                                    const int* __restrict__ cnt,
                                    const int* __restrict__ csrSrc,
                                    const float* __restrict__ dinv,
                                    const float* __restrict__ bias,
                                    const float* __restrict__ g,
                                    const float* __restrict__ be,
                                    const float* __restrict__ m,
                                    const float* __restrict__ v,
                                    float* __restrict__ out,
                                    int Nn) {
    int n = blockIdx.x * (blockDim.x >> 5) + (threadIdx.x >> 5);
    if (n >= Nn) return;
    const int lane = threadIdx.x & 31;
    const int c = lane << 2;
    const float di = dinv[n];
    const float sn = di * di;

    float4 hv = *(const float4*)(h + (size_t)n * 128 + c);
    float4 bv = *(const float4*)(bias + c);
    float ax = hv.x * sn + bv.x, ay = hv.y * sn + bv.y;
    float az = hv.z * sn + bv.z, aw = hv.w * sn + bv.w;

    const int s0 = rowStart[n];
    const int e0 = s0 + cnt[n];
    for (int i = s0; i < e0; ++i) {
        int s = csrSrc[i];
        if (i + 1 < e0)   // hide dependent random-row latency (global_prefetch_b8)
            __builtin_prefetch(h + (size_t)csrSrc[i + 1] * 128 + c, 0, 3);
        float nrm = dinv[s] * di;
        float4 r = *(const float4*)(h + (size_t)s * 128 + c);
        ax += r.x * nrm; ay += r.y * nrm; az += r.z * nrm; aw += r.w * nrm;
    }

    float4 gv = *(const float4*)(g + c);
    float4 bev = *(const float4*)(be + c);
    float4 mv = *(const float4*)(m + c);
    float4 vv = *(const float4*)(v + c);
    float4 o;
    o.x = fmaxf((ax - mv.x) * rsqrtf(vv.x + EPS) * gv.x + bev.x, 0.f);
    o.y = fmaxf((ay - mv.y) * rsqrtf(vv.y + EPS) * gv.y + bev.y, 0.f);
    o.z = fmaxf((az - mv.z) * rsqrtf(vv.z + EPS) * gv.z + bev.z, 0.f);
    o.w = fmaxf((aw - mv.w) * rsqrtf(vv.w + EPS) * gv.w + bev.w, 0.f);
    *(float4*)(out + (size_t)n * 128 + c) = o;
}

// ---------------------------------------------------------------------------
// Fused CSR gather + bias + BN + ReLU + residual, F=64 (lane -> float2 chunk).
// out = relu(bn(agg)) + x1[:, :64]
// ---------------------------------------------------------------------------
__global__ void gather_bn_relu_res_f64(const float* __restrict__ h,    // [N,64]
                                       const int* __restrict__ rowStart,
                                       const int* __restrict__ cnt,
                                       const int* __restrict__ csrSrc,
                                       const float* __restrict__ dinv,
                                       const float* __restrict__ bias,
                                       const float* __restrict__ g,
                                       const float* __restrict__ be,
                                       const float* __restrict__ m,
                                       const float* __restrict__ v,
                                       const float* __restrict__ x1,   // [N,128]
                                       float* __restrict__ out,        // [N,64]
                                       int Nn) {
    int n = blockIdx.x * (blockDim.x >> 5) + (threadIdx.x >> 5);
    if (n >= Nn) return;
    const int lane = threadIdx.x & 31;
    const int c = lane << 1;
    const float di = dinv[n];
    const float sn = di * di;

    float2 hv = *(const float2*)(h + (size_t)n * 64 + c);
    float ax = hv.x * sn + bias[c];
    float ay = hv.y * sn + bias[c + 1];

    const int s0 = rowStart[n];
    const int e0 = s0 + cnt[n];
    for (int i = s0; i < e0; ++i) {
        int s = csrSrc[i];
        if (i + 1 < e0)
            __builtin_prefetch(h + (size_t)csrSrc[i + 1] * 64 + c, 0, 3);
        float nrm = dinv[s] * di;
        float2 r = *(const float2*)(h + (size_t)s * 64 + c);
        ax += r.x * nrm; ay += r.y * nrm;
    }

    float2 res = *(const float2*)(x1 + (size_t)n * 128 + c);
    float2 o;
    o.x = fmaxf((ax - m[c])     * rsqrtf(v[c]     + EPS) * g[c]     + be[c],     0.f) + res.x;
    o.y = fmaxf((ay - m[c + 1]) * rsqrtf(v[c + 1] + EPS) * g[c + 1] + be[c + 1], 0.f) + res.y;
    *(float2*)(out + (size_t)n * 64 + c) = o;
}

// ---------------------------------------------------------------------------
// Layer-3 GEMM (64 -> 2): one wave per node, wave32 shuffle reduction
// ---------------------------------------------------------------------------
__global__ void gemm64x2_wave(const float* __restrict__ X,   // [N,64]
                              const float* __restrict__ W,   // [64,2]
                              float* __restrict__ H,         // [N,2]
                              int Nn) {
    int n = blockIdx.x * (blockDim.x >> 5) + (threadIdx.x >> 5);
    int lane = threadIdx.x & 31;
    if (n >= Nn) return;                     // uniform per wave
    const float2 xv = *(const float2*)(X + (size_t)n * 64 + lane * 2);
    float w00 = W[(lane * 2) * 2 + 0],     w01 = W[(lane * 2) * 2 + 1];
    float w10 = W[(lane * 2 + 1) * 2 + 0], w11 = W[(lane * 2 + 1) * 2 + 1];
    float p0 = xv.x * w00 + xv.y * w10;
    float p1 = xv.x * w01 + xv.y * w11;
    for (int off = 16; off > 0; off >>= 1) {
        p0 += __shfl_xor(p0, off, 32);
        p1 += __shfl_xor(p1, off, 32);
    }
    if (lane == 0) {
        H[(size_t)n * 2 + 0] = p0;
        H[(size_t)n * 2 + 1] = p1;
    }
}

// ---------------------------------------------------------------------------
// Layer-3 fused CSR gather + bias + log_softmax; lanes parallelize edges.
// ---------------------------------------------------------------------------
__global__ void gather_lsm_f2(const float* __restrict__ h3,   // [N,2]
                              const int* __restrict__ rowStart,
                              const int* __restrict__ cnt,
                              const int* __restrict__ csrSrc,
                              const float* __restrict__ dinv,
                              const float* __restrict__ b3,
                              float* __restrict__ out, int Nn) {
    int n = blockIdx.x * (blockDim.x >> 5) + (threadIdx.x >> 5);
    if (n >= Nn) return;                     // uniform per wave
    const int lane = threadIdx.x & 31;
    const float di = dinv[n];
    float p0 = 0.f, p1 = 0.f;
    const int s0 = rowStart[n];
    const int e0 = s0 + cnt[n];
    for (int i = s0 + lane; i < e0; i += 32) {
        int s = csrSrc[i];
        float nrm = dinv[s] * di;
        p0 += h3[(size_t)s * 2 + 0] * nrm;
        p1 += h3[(size_t)s * 2 + 1] * nrm;
    }
    for (int off = 16; off > 0; off >>= 1) {
        p0 += __shfl_xor(p0, off, 32);
        p1 += __shfl_xor(p1, off, 32);
    }
    if (lane == 0) {
        float sn = di * di;
        float a = p0 + h3[(size_t)n * 2 + 0] * sn + b3[0];
        float b = p1 + h3[(size_t)n * 2 + 1] * sn + b3[1];
        float mx = fmaxf(a, b);
        float l = mx + logf(expf(a - mx) + expf(b - mx));
        out[(size_t)n * 2 + 0] = a - l;
        out[(size_t)n * 2 + 1] = b - l;
    }
}

// ---------------------------------------------------------------------------
extern "C" void kernel_launch(void* const* d_in, const int* in_sizes, int n_in,
                              void* d_out, int out_size, void* d_ws, size_t ws_size,
                              hipStream_t stream) {
    const float* x   = (const float*)d_in[0];
    const int*   ei  = (const int*)d_in[1];
    const float* W1  = (const float*)d_in[2];
    const float* b1  = (const float*)d_in[3];
    const float* g1  = (const float*)d_in[4];
    const float* be1 = (const float*)d_in[5];
    const float* m1  = (const float*)d_in[6];
    const float* v1  = (const float*)d_in[7];
    const float* W2  = (const float*)d_in[8];
    const float* b2  = (const float*)d_in[9];
    const float* g2  = (const float*)d_in[10];
    const float* be2 = (const float*)d_in[11];
    const float* m2  = (const float*)d_in[12];
    const float* v2  = (const float*)d_in[13];
    const float* W3  = (const float*)d_in[14];
    const float* b3  = (const float*)d_in[15];
    float* out = (float*)d_out;

    const int N = in_sizes[0] / 512;
    const int E = in_sizes[1] / 2;
    const int* srcIdx = ei;
    const int* dstIdx = ei + E;

    // Workspace layout (all 4-byte elements; N, E multiples of 4 keep float4
    // alignment).  ints: cnt[N], rowStart[N], cursor[N], blockSums[512],
    // csrSrc[E]; floats: dinv[N], bufA[N*128], bufB[N*128].
    int* ip = (int*)d_ws;
    int* cnt       = ip;
    int* rowStart  = ip + N;
    int* cursor    = ip + 2 * (size_t)N;
    int* blockSums = ip + 3 * (size_t)N;
    int* csrSrc    = ip + 3 * (size_t)N + 512;
    float* dinv = (float*)(csrSrc + E);
    float* bufA = dinv + N;                              // h1 / h2 (+x_res)
    float* bufB = bufA + (size_t)N * 128;                // x1 / h3
    float* xres = bufA + (size_t)N * 64;                 // [N,64]

    const int B = 256;
    const int nb = (N + 255) / 256;                      // scan blocks (<=512)

    // --- CSR build + normalization ---
    fill_zero_i32<<<(N + B - 1) / B, B, 0, stream>>>(cnt, N);
    cnt_accum<<<(E + B - 1) / B, B, 0, stream>>>(dstIdx, cnt, E);
    dinv_from_cnt<<<(N + B - 1) / B, B, 0, stream>>>(cnt, dinv, N);
    scan_block<<<nb, 256, 0, stream>>>(cnt, rowStart, blockSums, N);
    scan_sums<<<1, 512, 0, stream>>>(blockSums, nb);
    scan_add_cursor<<<nb, 256, 0, stream>>>(rowStart, blockSums, cursor, N);
    csr_place<<<(E + B - 1) / B, B, 0, stream>>>(srcIdx, dstIdx, cursor, csrSrc, E);

    // --- layer 1: h1 = x @ W1 (512->128); fused gather+BN+ReLU -> x1 ---
    gcn_gemm_f32_wmma<<<(N + 31) / 32, 8 * 32, 32 * (512 + 4) * sizeof(float), stream>>>(
        x, W1, bufA, N, 512, 128);
    gather_bn_relu_f128<<<(N + 7) / 8, 256, 0, stream>>>(
        bufA, rowStart, cnt, csrSrc, dinv, b1, g1, be1, m1, v1, bufB, N);

    // --- layer 2: h2 = x1 @ W2 (128->64); fused gather+BN+ReLU+residual ---
    gcn_gemm_f32_wmma<<<(N + 31) / 32, 4 * 32, 32 * (128 + 4) * sizeof(float), stream>>>(
        bufB, W2, bufA, N, 128, 64);
    gather_bn_relu_res_f64<<<(N + 7) / 8, 256, 0, stream>>>(
        bufA, rowStart, cnt, csrSrc, dinv, b2, g2, be2, m2, v2, bufB, xres, N);

    // --- layer 3: h3 = x_res @ W3 (64->2); fused gather+bias+log_softmax ---
    gemm64x2_wave<<<(N + 7) / 8, 256, 0, stream>>>(xres, W3, bufB, N);
    gather_lsm_f2<<<(N + 7) / 8, 256, 0, stream>>>(
        bufB, rowStart, cnt, csrSrc, dinv, b3, out, N);
}